// EGATlayer_78151224918196
// MI455X (gfx1250) — compile-verified
//
#include <hip/hip_runtime.h>
#include <hip/hip_bf16.h>

#define N_NODES 100000
#define N_EDGES 1600000
#define IN_N 128
#define IN_E 64
#define NH 4            // heads
#define FE 16
#define FN 32
#define HFE (NH*FE)     // 64
#define HFN (NH*FN)     // 128
#define NPROJ (HFE + HFE + HFN)   // 256: [f_ni | f_nj | h]

#define K1_STRIDE 132   // 128 dwords + 4 pad -> bank = (4*r + k) % 64, clean
#define K2_STRIDE 68    // 64 dwords + 4 pad

typedef float v2f __attribute__((ext_vector_type(2)));
typedef float v8f __attribute__((ext_vector_type(8)));
typedef unsigned int v4u __attribute__((ext_vector_type(4)));
typedef int v4i __attribute__((ext_vector_type(4)));
typedef int v8i __attribute__((ext_vector_type(8)));

__device__ __forceinline__ v8f wmma4(v2f a, v2f b, v8f c) {
  // D = A(16x4) * B(4x16) + C, fp32 tensor op (V_WMMA_F32_16X16X4_F32)
  return __builtin_amdgcn_wmma_f32_16x16x4_f32(false, a, false, b, (short)0, c,
                                               false, false);
}

// ---- Tensor Data Mover: 2D fp32 tile Global->LDS with LDS row padding ----
// rows x cols (dwords), source row pitch = pitch dwords; LDS gets pad_amount
// (code) dwords inserted every 2^(interval_code+1) dwords.
__device__ __forceinline__ void tdm_load_2d(const void* gptr, unsigned lds_off,
                                            unsigned rows, unsigned cols,
                                            unsigned pitch,
                                            unsigned interval_code,
                                            unsigned amount_code) {
  unsigned long long ga = (unsigned long long)(uintptr_t)gptr;
  v4u g0;
  g0.x = 1u;                                        // count=1 (valid user D#)
  g0.y = lds_off;                                   // lds_addr (bytes)
  g0.z = (unsigned)(ga & 0xFFFFFFFFu);              // global_addr[31:0]
  g0.w = (unsigned)((ga >> 32) & 0x01FFFFFFu)       // global_addr[56:32]
       | (2u << 30);                                // type = 2 (image)
  v8i g1;
  g1[0] = (int)((2u << 16)                          // data_size = 4B
              | (1u << 20)                          // pad_enable
              | (interval_code << 22)               // pad_interval
              | (amount_code << 25));               // pad_amount
  g1[1] = (int)((cols & 0xFFFFu) << 16);            // tensor_dim0 lo16
  g1[2] = (int)(((cols >> 16) & 0xFFFFu)            // tensor_dim0 hi16
              | ((rows & 0xFFFFu) << 16));          // tensor_dim1 lo16
  g1[3] = (int)(((rows >> 16) & 0xFFFFu)            // tensor_dim1 hi16
              | ((cols & 0xFFFFu) << 16));          // tile_dim0
  g1[4] = (int)(rows & 0xFFFFu);                    // tile_dim1 (tile_dim2=0)
  g1[5] = (int)pitch;                               // tensor_dim0_stride lo32
  g1[6] = 0;
  g1[7] = 0;
  v4i z4 = {0, 0, 0, 0};
#if __has_include(<hip/amd_detail/amd_gfx1250_TDM.h>)
  v8i z8 = {0, 0, 0, 0, 0, 0, 0, 0};
  __builtin_amdgcn_tensor_load_to_lds(g0, g1, z4, z4, z8, 0);
#else
  __builtin_amdgcn_tensor_load_to_lds(g0, g1, z4, z4, 0);
#endif
}

// ordered-uint encoding so unsigned atomicMax == float max
__device__ __forceinline__ unsigned f2key(float f) {
  unsigned u = __float_as_uint(f);
  return (u & 0x80000000u) ? ~u : (u | 0x80000000u);
}
__device__ __forceinline__ float key2f(unsigned k) {
  return (k & 0x80000000u) ? __uint_as_float(k & 0x7fffffffu)
                           : __uint_as_float(~k);
}

// ---------------- K0: init outputs / stats ----------------
__global__ void k0_init(float* h_out, unsigned* emax, float* denom) {
  long long i = (long long)blockIdx.x * blockDim.x + threadIdx.x;
  if (i < (long long)N_NODES * HFN) h_out[i] = 0.0f;
  if (i < (long long)N_NODES * NH) { emax[i] = 0u; denom[i] = 0.0f; }
}

// -------- K1: node_proj = nfeats @ [W_ni | W_nj | W_node] (WMMA + TDM) -----
__global__ void __launch_bounds__(128)
k1_node_proj(const float* __restrict__ nf,
             const float* __restrict__ Wni, const float* __restrict__ Wnj,
             const float* __restrict__ Wnode, const float* __restrict__ bnode,
             float* __restrict__ proj) {
  __shared__ float wt[64 * K1_STRIDE];        // transposed weight slice
  __shared__ float at[4][16 * K1_STRIDE];     // per-wave A tiles (TDM dest)

  int g = blockIdx.y;                          // 64-col group of the 256 cols
  const float* W; int wn; int c0;
  if (g == 0)      { W = Wni;   wn = HFE; c0 = 0;  }
  else if (g == 1) { W = Wnj;   wn = HFE; c0 = 0;  }
  else if (g == 2) { W = Wnode; wn = HFN; c0 = 0;  }
  else             { W = Wnode; wn = HFN; c0 = 64; }

  // stage weights transposed: wt[col][k] so B fragments are ds_load_b64
  for (int i = threadIdx.x; i < IN_N * 64; i += blockDim.x) {
    int k = i >> 6, col = i & 63;
    wt[col * K1_STRIDE + k] = W[k * wn + c0 + col];
  }

  int wv = __builtin_amdgcn_readfirstlane((int)(threadIdx.x >> 5));
  int mt = blockIdx.x * 4 + wv;                // node tile (16 nodes)
  bool active = (mt < N_NODES / 16);
  if (active) {                                // async DMA of 16x128 A tile
    tdm_load_2d(nf + (long long)mt * 16 * IN_N,
                (unsigned)(uintptr_t)&at[wv][0], 16, IN_N, IN_N,
                /*every 128 dwords*/ 6, /*pad 4 dwords*/ 3);
  }
  __syncthreads();
  if (!active) return;
  __builtin_amdgcn_s_wait_tensorcnt((short)0);
  asm volatile("" ::: "memory");

  int lane = threadIdx.x & 31;
  int half = lane >> 4;
  int l15  = lane & 15;
  int m0 = mt * 16;

  v8f acc[4];
  const v8f vzero = {0.f,0.f,0.f,0.f,0.f,0.f,0.f,0.f};
  for (int j = 0; j < 4; ++j) acc[j] = vzero;

  for (int s = 0; s < IN_N / 4; ++s) {
    int k = 4 * s + 2 * half;                  // lanes 0-15: K0,1; 16-31: K2,3
    v2f a = *(const v2f*)&at[wv][l15 * K1_STRIDE + k];
#pragma unroll
    for (int j = 0; j < 4; ++j) {
      v2f b = *(const v2f*)&wt[(16 * j + l15) * K1_STRIDE + k];
      acc[j] = wmma4(a, b, acc[j]);
    }
  }
#pragma unroll
  for (int j = 0; j < 4; ++j) {
    int colg = 64 * g + 16 * j + l15;          // column in [0,256)
    float bb = (colg >= 2 * HFE) ? bnode[colg - 2 * HFE] : 0.0f;
#pragma unroll
    for (int v = 0; v < 8; ++v) {
      int row = m0 + v + 8 * half;             // C layout: M = v / v+8
      proj[(long long)row * NPROJ + colg] = acc[j][v] + bb;
    }
  }
}

// -------- K2: edge GEMM (TDM-fed) + gather + leaky + logits + segmax ------
__global__ void __launch_bounds__(128)
k2_edge(const float* __restrict__ ef, const int* __restrict__ src,
        const int* __restrict__ dst, const float* __restrict__ Wfij,
        const float* __restrict__ attn, const float* __restrict__ bias,
        const float* __restrict__ proj, float* __restrict__ fout,
        float* __restrict__ e_ws, unsigned* __restrict__ emax) {
  __shared__ float wt[64 * K2_STRIDE];         // transposed W_fij [col][k]
  __shared__ float at[4][16 * K2_STRIDE];      // per-wave 16x64 edge tiles

  for (int i = threadIdx.x; i < IN_E * HFE; i += blockDim.x) {
    int k = i >> 6, col = i & 63;
    wt[col * K2_STRIDE + k] = Wfij[k * HFE + col];
  }

  int wv = __builtin_amdgcn_readfirstlane((int)(threadIdx.x >> 5));
  int tile = blockIdx.x * 4 + wv;              // exact: E/16 = 100000 tiles
  tdm_load_2d(ef + (long long)tile * 16 * IN_E,
              (unsigned)(uintptr_t)&at[wv][0], 16, IN_E, IN_E,
              /*every 64 dwords*/ 5, /*pad 4 dwords*/ 3);
  __syncthreads();
  __builtin_amdgcn_s_wait_tensorcnt((short)0);
  asm volatile("" ::: "memory");

  int lane = threadIdx.x & 31;
  int half = lane >> 4;
  int l15 = lane & 15;
  int e0 = tile * 16;

  v8f acc[4];
  const v8f vzero = {0.f,0.f,0.f,0.f,0.f,0.f,0.f,0.f};
  for (int j = 0; j < 4; ++j) acc[j] = vzero;

  for (int s = 0; s < IN_E / 4; ++s) {
    int k = 4 * s + 2 * half;
    v2f a = *(const v2f*)&at[wv][l15 * K2_STRIDE + k];
#pragma unroll
    for (int j = 0; j < 4; ++j) {
      v2f b = *(const v2f*)&wt[(16 * j + l15) * K2_STRIDE + k];
      acc[j] = wmma4(a, b, acc[j]);
    }
  }

  int my_src = src[e0 + l15];
  int my_dst = dst[e0 + l15];

#pragma unroll
  for (int j = 0; j < 4; ++j) {                // j == head (FE == tile width)
    int col = 16 * j + l15;
    float at_w = attn[col];
    float bi = bias[col];
#pragma unroll
    for (int v = 0; v < 8; ++v) {
      int row = v + 8 * half;
      int se = __shfl(my_src, row, 32);
      int de = __shfl(my_dst, row, 32);
      float c = acc[j][v]
              + proj[(long long)se * NPROJ + col]          // f_ni[src]
              + proj[(long long)de * NPROJ + HFE + col]    // f_nj[dst]
              + bi;
      c = (c > 0.0f) ? c : 0.01f * c;                      // leaky_relu
      long long edge = e0 + row;
      fout[edge * HFE + col] = c;                          // output f_out
      float wvv = c * at_w;                                // logit partial
#pragma unroll
      for (int off = 1; off < 16; off <<= 1)               // 16-lane reduce
        wvv += __shfl_xor(wvv, off, 32);
      if (l15 == 0) {
        e_ws[edge * NH + j] = wvv;
        atomicMax(&emax[(long long)de * NH + j], f2key(wvv));
      }
    }
  }
}

// ---------------- K3: a = exp(e - max), denom += a ------------------------
__global__ void k3_softmax(const int* __restrict__ dst,
                           const unsigned* __restrict__ emax,
                           float* __restrict__ ea, float* __restrict__ denom) {
  long long i = (long long)blockIdx.x * blockDim.x + threadIdx.x;
  if (i >= (long long)N_EDGES * NH) return;
  int edge = (int)(i >> 2);
  int h = (int)(i & 3);
  int d = dst[edge];
  float m = key2f(emax[(long long)d * NH + h]);
  float a = __expf(ea[i] - m);
  ea[i] = a;
  atomicAdd(&denom[(long long)d * NH + h], a);
}

// ---------------- K4: h_out[dst] += h[src] * a/denom ----------------------
__global__ void k4_aggregate(const int* __restrict__ src,
                             const int* __restrict__ dst,
                             const float* __restrict__ proj,
                             const float* __restrict__ a_ws,
                             const float* __restrict__ denom,
                             float* __restrict__ h_out) {
  long long i = (long long)blockIdx.x * blockDim.x + threadIdx.x;
  if (i >= (long long)N_EDGES * HFN) return;
  int edge = (int)(i >> 7);
  int c = (int)(i & 127);
  int h = c >> 5;                              // FN=32 per head
  int s = src[edge];
  int d = dst[edge];
  float a = a_ws[(long long)edge * NH + h] / denom[(long long)d * NH + h];
  float val = proj[(long long)s * NPROJ + 2 * HFE + c] * a;
  atomicAdd(&h_out[(long long)d * HFN + c], val);
}

extern "C" void kernel_launch(void* const* d_in, const int* in_sizes, int n_in,
                              void* d_out, int out_size, void* d_ws,
                              size_t ws_size, hipStream_t stream) {
  const float* nfeats = (const float*)d_in[0];
  const float* efeats = (const float*)d_in[1];
  const int*   src    = (const int*)d_in[2];
  const int*   dst    = (const int*)d_in[3];
  const float* W_node = (const float*)d_in[4];
  const float* b_node = (const float*)d_in[5];
  const float* W_ni   = (const float*)d_in[6];
  const float* W_nj   = (const float*)d_in[7];
  const float* W_fij  = (const float*)d_in[8];
  const float* attn   = (const float*)d_in[9];
  const float* bias   = (const float*)d_in[10];

  float* h_out = (float*)d_out;                                 // [N,H,FN]
  float* f_out = (float*)d_out + (long long)N_NODES * HFN;      // [E,H,FE]

  char* ws = (char*)d_ws;
  float* proj = (float*)ws;       ws += (size_t)N_NODES * NPROJ * sizeof(float);
  float* ea   = (float*)ws;       ws += (size_t)N_EDGES * NH * sizeof(float);
  unsigned* emax = (unsigned*)ws; ws += (size_t)N_NODES * NH * sizeof(unsigned);
  float* denom = (float*)ws;

  {
    long long total = (long long)N_NODES * HFN;
    k0_init<<<(int)((total + 255) / 256), 256, 0, stream>>>(h_out, emax, denom);
  }
  {
    int mtiles = N_NODES / 16;                 // 6250
    dim3 grid((mtiles + 3) / 4, 4);            // 1563 x 4 weight groups
    k1_node_proj<<<grid, 128, 0, stream>>>(nfeats, W_ni, W_nj, W_node, b_node,
                                           proj);
  }
  {
    int tiles = N_EDGES / 16;                  // 100000, 4 waves per block
    k2_edge<<<tiles / 4, 128, 0, stream>>>(efeats, src, dst, W_fij, attn, bias,
                                           proj, f_out, ea, emax);
  }
  {
    long long total = (long long)N_EDGES * NH;
    k3_softmax<<<(int)((total + 255) / 256), 256, 0, stream>>>(dst, emax, ea,
                                                               denom);
  }
  {
    long long total = (long long)N_EDGES * HFN;
    k4_aggregate<<<(int)((total + 255) / 256), 256, 0, stream>>>(src, dst, proj,
                                                                 ea, denom,
                                                                 h_out);
  }
}